// SpectralConvS2_1090921693676
// MI455X (gfx1250) — compile-verified
//
#include <hip/hip_runtime.h>
#include <hip/hip_bf16.h>
#include <stddef.h>

// ---------------------------------------------------------------------------
// SpectralConvS2 on MI455X (gfx1250, wave32).
// y = irfft2( einsum("bilm,oil->bolm", rfft2(x), W) ), residual = x.
// Bandwidth-bound (~1.6-2 GB HBM traffic vs ~21.5 GFLOP), so fp32 everywhere;
// per-l complex 32x32 channel mix via V_WMMA_F32_16X16X4_F32; strided lat-FFT
// tiles moved by the Tensor Data Mover (TENSOR_LOAD_TO_LDS / STORE_FROM_LDS).
// ---------------------------------------------------------------------------

#define NLAT 720
#define NLON 1440
#define NM   721              // NLON/2 + 1
#define CCH  32               // C_IN == C_OUT
#define NB   2

#define LATCG 4                               // lat-FFT columns per workgroup
#define LATMT ((NM + LATCG - 1) / LATCG)      // 181 m-tiles

typedef __attribute__((ext_vector_type(2))) float v2f;
typedef __attribute__((ext_vector_type(8))) float v8f;

#define TWO_PI 6.28318530717958647692f

// ---------------- Tensor Data Mover helpers --------------------------------
#if defined(__has_builtin)
#  if __has_builtin(__builtin_amdgcn_tensor_load_to_lds) && \
      __has_builtin(__builtin_amdgcn_tensor_store_from_lds)
#    define HAVE_TDM 1
#  endif
#endif
#ifndef HAVE_TDM
#  define HAVE_TDM 0
#endif

#if HAVE_TDM
typedef __attribute__((ext_vector_type(4))) unsigned int tdm_v4u;
typedef __attribute__((ext_vector_type(8))) int          tdm_v8i;
typedef __attribute__((ext_vector_type(4))) int          tdm_v4i;

// Build a 2D tile D# (groups 0/1) per CDNA5 ISA §8.3/8.4.
// data_size = 8B (complex64). tile = tile0 x tile1, tensor dim0 stride given
// in elements. OOB columns (x >= tdim0) read zero / drop stores.
__device__ __forceinline__ void tdm_desc_2d(tdm_v4u& g0, tdm_v8i& g1,
                                            unsigned lds_addr,
                                            unsigned long long gaddr,
                                            unsigned tdim0, unsigned tdim1,
                                            unsigned tile0, unsigned tile1,
                                            unsigned long long stride0)
{
    g0[0] = 1u;                                              // count=1, user desc
    g0[1] = lds_addr;                                        // lds_addr [63:32]
    g0[2] = (unsigned)gaddr;                                 // global_addr lo
    g0[3] = (unsigned)((gaddr >> 32) & 0x1FFFFFFull)         // global_addr hi (57b)
          | (2u << 30);                                      // type=2 ("image")
    g1[0] = (int)(3u << 16);                                 // data_size=3 (8B)
    g1[1] = (int)((tdim0 & 0xFFFFu) << 16);                  // tensor_dim0 lo16
    g1[2] = (int)((tdim0 >> 16) | ((tdim1 & 0xFFFFu) << 16));// dim0 hi / dim1 lo
    g1[3] = (int)((tdim1 >> 16) | ((tile0 & 0xFFFFu) << 16));// dim1 hi / tile_dim0
    g1[4] = (int)(tile1 & 0xFFFFu);                          // tile_dim1 (tile2=0)
    g1[5] = (int)(unsigned)(stride0 & 0xFFFFFFFFull);        // dim0_stride lo32
    g1[6] = (int)(unsigned)((stride0 >> 32) & 0xFFFFull);    // dim0_stride hi16
    g1[7] = 0;
}

__device__ __forceinline__ void tdm_load(const tdm_v4u& g0, const tdm_v8i& g1)
{
    tdm_v4i z4 = {};
#if defined(__clang_major__) && (__clang_major__ >= 23)
    tdm_v8i z8 = {};
    __builtin_amdgcn_tensor_load_to_lds(g0, g1, z4, z4, z8, 0);
#else
    __builtin_amdgcn_tensor_load_to_lds(g0, g1, z4, z4, 0);
#endif
}

__device__ __forceinline__ void tdm_store(const tdm_v4u& g0, const tdm_v8i& g1)
{
    tdm_v4i z4 = {};
#if defined(__clang_major__) && (__clang_major__ >= 23)
    tdm_v8i z8 = {};
    __builtin_amdgcn_tensor_store_from_lds(g0, g1, z4, z4, z8, 0);
#else
    __builtin_amdgcn_tensor_store_from_lds(g0, g1, z4, z4, 0);
#endif
}
#endif // HAVE_TDM

// ---------------- Stockham mixed-radix FFT in LDS ---------------------------
// S = LDS element stride (columns interleaved), c0 = column offset.
template<int R, int S>
__device__ __forceinline__ void fft_stage(const float2* __restrict__ src,
                                          float2* __restrict__ dst,
                                          int N, int Ns, float sign,
                                          int c0, int tid, int nt)
{
    const int NR = N / R;
    for (int j = tid; j < NR; j += nt) {
        float2 v[R];
        const float ang0 = sign * TWO_PI * (float)(j % Ns) / (float)(Ns * R);
        #pragma unroll
        for (int r = 0; r < R; ++r) {
            float2 xv = src[(j + r * NR) * S + c0];
            float sa, ca;
            __sincosf(ang0 * (float)r, &sa, &ca);
            v[r] = make_float2(xv.x * ca - xv.y * sa, xv.x * sa + xv.y * ca);
        }
        const int idxD = (j / Ns) * (Ns * R) + (j % Ns);
        #pragma unroll
        for (int k = 0; k < R; ++k) {
            float2 acc = v[0];
            #pragma unroll
            for (int r = 1; r < R; ++r) {
                float sa, ca;
                __sincosf(sign * TWO_PI * (float)((r * k) % R) / (float)R, &sa, &ca);
                acc.x += v[r].x * ca - v[r].y * sa;
                acc.y += v[r].x * sa + v[r].y * ca;
            }
            dst[(idxD + k * Ns) * S + c0] = acc;
        }
    }
    __syncthreads();
}

// 4-factor FFT: result lands back in s0 (even number of ping-pongs).
template<int N, int R0, int R1, int R2, int R3, int SGN, int S>
__device__ __forceinline__ float2* fft4(float2* s0, float2* s1,
                                        int c0, int tid, int nt)
{
    const float sign = (float)SGN;
    int Ns = 1;
    fft_stage<R0, S>(s0, s1, N, Ns, sign, c0, tid, nt); Ns *= R0;
    fft_stage<R1, S>(s1, s0, N, Ns, sign, c0, tid, nt); Ns *= R1;
    fft_stage<R2, S>(s0, s1, N, Ns, sign, c0, tid, nt); Ns *= R2;
    fft_stage<R3, S>(s1, s0, N, Ns, sign, c0, tid, nt);
    return s0;
}

// ---------------- Kernel 1: rfft along lon (1440 = 8*6*6*5) -----------------
__global__ void k_lon_fwd(const float* __restrict__ x, float2* __restrict__ out)
{
    __shared__ float2 s0[NLON];
    __shared__ float2 s1[NLON];
    const int row = blockIdx.x;                 // (b*32+c)*720 + lat
    const float* __restrict__ xr = x + (size_t)row * NLON;

    for (int i = threadIdx.x; i < NLON; i += blockDim.x)
        s0[i] = make_float2(xr[i], 0.0f);
    __syncthreads();

    float2* res = fft4<NLON, 8, 6, 6, 5, -1, 1>(s0, s1, 0, threadIdx.x, blockDim.x);

    const float sc = 0.02635231383473649f;      // 1/sqrt(1440)
    float2* __restrict__ o = out + (size_t)row * NM;
    for (int m = threadIdx.x; m < NM; m += blockDim.x) {
        float2 v = res[m];
        o[m] = make_float2(v.x * sc, v.y * sc);
    }
}

// ---------------- Kernel 2/4: complex FFT along lat (720 = 6*6*4*5) ---------
// LATCG columns per workgroup. The [720 x LATCG] strided tile (row stride
// 721 complex) is fetched/written by the Tensor Data Mover; columns are
// interleaved in LDS (stride LATCG) and 64 threads run each column FFT.
template<int SGN>
__global__ void k_lat(const float2* __restrict__ in, float2* __restrict__ out)
{
    __shared__ float2 s0[NLAT * LATCG];
    __shared__ float2 s1[NLAT * LATCG];
    const int bc = blockIdx.x / LATMT;
    const int mt = blockIdx.x % LATMT;
    const int m0 = mt * LATCG;
    const size_t gbase = (size_t)bc * NLAT * NM + m0;   // complex elements

#if HAVE_TDM
    if (threadIdx.x < 32) {                      // wave 0 owns the DMA
        tdm_v4u g0; tdm_v8i g1;
        tdm_desc_2d(g0, g1,
                    (unsigned)(size_t)(void*)s0,
                    (unsigned long long)(size_t)(const void*)(in + gbase),
                    (unsigned)(NM - m0), NLAT,   // tensor dims (OOB -> zero)
                    LATCG, NLAT,                 // tile dims
                    NM);                         // row stride (elements)
        tdm_load(g0, g1);
        __builtin_amdgcn_s_wait_tensorcnt(0);
    }
    __syncthreads();
#else
    for (int idx = threadIdx.x; idx < NLAT * LATCG; idx += blockDim.x) {
        const int lat = idx / LATCG, mm = idx % LATCG;
        s0[idx] = (m0 + mm < NM) ? in[gbase + (size_t)lat * NM + mm]
                                 : make_float2(0.f, 0.f);
    }
    __syncthreads();
#endif

    const int col = threadIdx.x & (LATCG - 1);
    const int tid = threadIdx.x / LATCG;         // 0..63 per column
    float2* res = fft4<NLAT, 6, 6, 4, 5, SGN, LATCG>(s0, s1, col, tid,
                                                     blockDim.x / LATCG);

    const float sc = 0.0372677996249965f;        // 1/sqrt(720), ortho both ways
    for (int idx = threadIdx.x; idx < NLAT * LATCG; idx += blockDim.x) {
        float2 v = res[idx];
        res[idx] = make_float2(v.x * sc, v.y * sc);
    }
    __syncthreads();

#if HAVE_TDM
    if (threadIdx.x < 32) {
        tdm_v4u g0; tdm_v8i g1;
        tdm_desc_2d(g0, g1,
                    (unsigned)(size_t)(void*)res,
                    (unsigned long long)(size_t)(void*)(out + gbase),
                    (unsigned)(NM - m0), NLAT,   // OOB columns dropped
                    LATCG, NLAT,
                    NM);
        tdm_store(g0, g1);
        __builtin_amdgcn_s_wait_tensorcnt(0);    // hold LDS until DMA done
    }
#else
    for (int idx = threadIdx.x; idx < NLAT * LATCG; idx += blockDim.x) {
        const int lat = idx / LATCG, mm = idx % LATCG;
        if (m0 + mm < NM)
            out[gbase + (size_t)lat * NM + mm] = res[idx];
    }
#endif
}

// ---------------- Kernel 3: per-l complex channel mix via fp32 WMMA ---------
// yh[b,o,l,m] = sum_i W[o,i,l] * xh[b,i,l,m]   (complex)
// Per wave: one 16(o) x 16(m) tile, K=32 in chunks of 4 via
// V_WMMA_F32_16X16X4_F32. -Wi*Xi uses a pre-negated A fragment (f32 WMMA has
// only C-neg). A layout: lane%16=M, lane/16 selects K pair. D: lane%16=N,
// VGPR r -> M = r + 8*(lane/16).
__global__ void k_mix(const float2* __restrict__ xh,
                      const float2* __restrict__ w,     // weight[o][i][l]
                      float2* __restrict__ yh)
{
    const int wave = threadIdx.x >> 5;
    const int lane = threadIdx.x & 31;
    const int t  = blockIdx.x * 8 + wave;       // 132480 tiles total
    const int mt = t % 46;
    const int ot = (t / 46) & 1;
    const int l  = (t / 92) % NLAT;
    const int b  = t / (92 * NLAT);

    const int hf = lane >> 4;
    const int q  = lane & 15;

    const int m   = mt * 16 + q;
    const bool mok = (m < NM);
    const int mv  = mok ? m : (NM - 1);
    const int oM  = ot * 16 + q;

    v8f cr = {};
    v8f ci = {};

    #pragma unroll
    for (int kk = 0; kk < CCH; kk += 4) {
        const int i0 = kk + hf * 2;
        const int i1 = i0 + 1;

        // Prefetch next chunk's B row (global_prefetch_b8).
        if (kk + 4 < CCH)
            __builtin_prefetch(&xh[((size_t)(b * CCH + kk + 4) * NLAT + l) * NM + mv], 0, 1);

        float2 w0 = w[(size_t)(oM * CCH + i0) * NLAT + l];
        float2 w1 = w[(size_t)(oM * CCH + i1) * NLAT + l];
        v2f ar = {w0.x, w1.x};
        v2f ai = {w0.y, w1.y};
        v2f an = {-w0.y, -w1.y};

        float2 x0 = xh[((size_t)(b * CCH + i0) * NLAT + l) * NM + mv];
        float2 x1 = xh[((size_t)(b * CCH + i1) * NLAT + l) * NM + mv];
        if (!mok) { x0 = make_float2(0.f, 0.f); x1 = make_float2(0.f, 0.f); }
        v2f br = {x0.x, x1.x};
        v2f bi = {x0.y, x1.y};

        cr = __builtin_amdgcn_wmma_f32_16x16x4_f32(false, ar, false, br, (short)0, cr, false, false);
        cr = __builtin_amdgcn_wmma_f32_16x16x4_f32(false, an, false, bi, (short)0, cr, false, false);
        ci = __builtin_amdgcn_wmma_f32_16x16x4_f32(false, ar, false, bi, (short)0, ci, false, false);
        ci = __builtin_amdgcn_wmma_f32_16x16x4_f32(false, ai, false, br, (short)0, ci, false, false);
    }

    if (mok) {
        #pragma unroll
        for (int r = 0; r < 8; ++r) {
            const int o = ot * 16 + r + 8 * hf;
            yh[((size_t)(b * CCH + o) * NLAT + l) * NM + m] = make_float2(cr[r], ci[r]);
        }
    }
}

// ---------------- Kernel 5: irfft along lon --------------------------------
__global__ void k_lon_inv(const float2* __restrict__ yh, float* __restrict__ y)
{
    __shared__ float2 s0[NLON];
    __shared__ float2 s1[NLON];
    const int row = blockIdx.x;                 // (b*32+o)*720 + lat
    const float2* __restrict__ src = yh + (size_t)row * NM;

    for (int m = threadIdx.x; m < NM; m += blockDim.x)
        s0[m] = src[m];
    for (int m = NM + threadIdx.x; m < NLON; m += blockDim.x) {
        float2 v = src[NLON - m];               // Hermitian extension
        s0[m] = make_float2(v.x, -v.y);
    }
    __syncthreads();

    float2* res = fft4<NLON, 8, 6, 6, 5, +1, 1>(s0, s1, 0, threadIdx.x, blockDim.x);

    const float sc = 0.02635231383473649f;      // 1/sqrt(1440)
    float* __restrict__ yr = y + (size_t)row * NLON;
    for (int i = threadIdx.x; i < NLON; i += blockDim.x)
        yr[i] = res[i].x * sc;
}

// ---------------- Residual copy (float4, BW-limited) ------------------------
__global__ void k_copy4(const float4* __restrict__ in, float4* __restrict__ out, int n)
{
    const int i = blockIdx.x * blockDim.x + threadIdx.x;
    if (i < n) out[i] = in[i];
}

// ---------------------------------------------------------------------------
extern "C" void kernel_launch(void* const* d_in, const int* in_sizes, int n_in,
                              void* d_out, int out_size, void* d_ws, size_t ws_size,
                              hipStream_t stream)
{
    const float*  x   = (const float*)d_in[0];        // [2,32,720,1440] f32
    const float2* wgt = (const float2*)d_in[1];       // [32,32,720] complex64
    float* out = (float*)d_out;                       // y ++ residual

    const size_t nspec = (size_t)NB * CCH * NLAT * NM;   // 33,223,680 complex
    float2* bufA = (float2*)d_ws;
    float2* bufB = bufA + nspec;

    const int rows     = NB * CCH * NLAT;             // 46080
    const int lat_blks = NB * CCH * LATMT;            // 64 * 181 = 11584

    k_lon_fwd<<<rows, 256, 0, stream>>>(x, bufA);
    k_lat<-1><<<lat_blks, 256, 0, stream>>>(bufA, bufB);
    k_mix<<<16560, 256, 0, stream>>>(bufB, wgt, bufA);   // 16560*8 = 132480 tiles
    k_lat<+1><<<lat_blks, 256, 0, stream>>>(bufA, bufB);
    k_lon_inv<<<rows, 256, 0, stream>>>(bufB, out);

    const size_t ysz = (size_t)NB * CCH * NLAT * NLON;   // 66,355,200 floats
    const int n4 = (int)(ysz / 4);
    k_copy4<<<(n4 + 255) / 256, 256, 0, stream>>>((const float4*)x,
                                                  (float4*)(out + ysz), n4);
}